// UniGCNIIConv_67688684585239
// MI455X (gfx1250) — compile-verified
//
#include <hip/hip_runtime.h>

#define FD 128
#define LN_EPS 1e-5f

typedef float v2f __attribute__((ext_vector_type(2)));
typedef float v8f __attribute__((ext_vector_type(8)));

// ---------------- K0: zero scratch ----------------
__global__ void zero_kernel(float* p, size_t n) {
    size_t i = (size_t)blockIdx.x * blockDim.x + threadIdx.x;
    size_t stride = (size_t)gridDim.x * blockDim.x;
    for (; i < n; i += stride) p[i] = 0.0f;
}

// ---------------- K1: X[vertex] scatter-add into esum[edges], count edges ----------------
__global__ __launch_bounds__(256) void scatter_v2e(const float* __restrict__ X,
                                                   const int* __restrict__ vertex,
                                                   const int* __restrict__ edges,
                                                   float* __restrict__ esum,
                                                   float* __restrict__ ecnt, int M_) {
    int t = blockIdx.x * 256 + threadIdx.x;
    int m = t >> 5;              // pair index
    if (m >= M_) return;
    int c = t & 31;              // 32 chunks of 4 floats = 128 features
    int v = vertex[m];
    int e = edges[m];
    const float4 x = *(const float4*)(X + (size_t)v * FD + c * 4);
    float* dst = esum + (size_t)e * FD + c * 4;
    atomicAdd(dst + 0, x.x);
    atomicAdd(dst + 1, x.y);
    atomicAdd(dst + 2, x.z);
    atomicAdd(dst + 3, x.w);
    if (c == 0) atomicAdd(ecnt + e, 1.0f);
}

// ---------------- K2: Xe = (esum / max(cnt,1)) * degE ----------------
__global__ __launch_bounds__(256) void edge_finalize(const float* __restrict__ esum,
                                                     const float* __restrict__ ecnt,
                                                     const float* __restrict__ degE,
                                                     float* __restrict__ Xe, int E_) {
    int t = blockIdx.x * 256 + threadIdx.x;
    int e = t >> 5;
    if (e >= E_) return;
    int c = t & 31;
    float s = degE[e] / fmaxf(ecnt[e], 1.0f);
    const float4 in = *(const float4*)(esum + (size_t)e * FD + c * 4);
    float4 o = {in.x * s, in.y * s, in.z * s, in.w * s};
    *(float4*)(Xe + (size_t)e * FD + c * 4) = o;
}

// ---------------- K3: Xe[edges] scatter-add into Xv[vertex] ----------------
__global__ __launch_bounds__(256) void scatter_e2v(const float* __restrict__ Xe,
                                                   const int* __restrict__ vertex,
                                                   const int* __restrict__ edges,
                                                   float* __restrict__ Xv, int M_) {
    int t = blockIdx.x * 256 + threadIdx.x;
    int m = t >> 5;
    if (m >= M_) return;
    int c = t & 31;
    int v = vertex[m];
    int e = edges[m];
    const float4 x = *(const float4*)(Xe + (size_t)e * FD + c * 4);
    float* dst = Xv + (size_t)v * FD + c * 4;
    atomicAdd(dst + 0, x.x);
    atomicAdd(dst + 1, x.y);
    atomicAdd(dst + 2, x.z);
    atomicAdd(dst + 3, x.w);
}

// ---------------- K4: degV scale, x2 residual, LayerNorm, GCNII alpha mix -> Xi ----------------
// one wave32 per node, 4 floats per lane
__global__ __launch_bounds__(256) void node_ln(const float* __restrict__ Xv,
                                               const float* __restrict__ degV,
                                               const float* __restrict__ X0,
                                               const float* __restrict__ ln_w,
                                               const float* __restrict__ ln_b,
                                               const float* __restrict__ alpha_p,
                                               float* __restrict__ Xi, int N_) {
    int warp = threadIdx.x >> 5;
    int lane = threadIdx.x & 31;
    int node = blockIdx.x * 8 + warp;
    if (node >= N_) return;
    const float alpha = *alpha_p;
    const float s = 2.0f * degV[node];   // Xh = Xv + Xv, Xv = segsum * degV
    const float4 xv = *(const float4*)(Xv + (size_t)node * FD + lane * 4);
    float4 xh = {xv.x * s, xv.y * s, xv.z * s, xv.w * s};
    float sum = xh.x + xh.y + xh.z + xh.w;
    float sq  = xh.x * xh.x + xh.y * xh.y + xh.z * xh.z + xh.w * xh.w;
    #pragma unroll
    for (int off = 16; off > 0; off >>= 1) {
        sum += __shfl_xor(sum, off, 32);
        sq  += __shfl_xor(sq,  off, 32);
    }
    const float mu   = sum * (1.0f / FD);
    const float var  = sq * (1.0f / FD) - mu * mu;
    const float rstd = rsqrtf(var + LN_EPS);
    const float4 w  = *(const float4*)(ln_w + lane * 4);
    const float4 b  = *(const float4*)(ln_b + lane * 4);
    const float4 x0 = *(const float4*)(X0 + (size_t)node * FD + lane * 4);
    const float oma = 1.0f - alpha;
    float4 o;
    o.x = oma * ((xh.x - mu) * rstd * w.x + b.x) + alpha * x0.x;
    o.y = oma * ((xh.y - mu) * rstd * w.y + b.y) + alpha * x0.y;
    o.z = oma * ((xh.z - mu) * rstd * w.z + b.z) + alpha * x0.z;
    o.w = oma * ((xh.w - mu) * rstd * w.w + b.w) + alpha * x0.w;
    *(float4*)(Xi + (size_t)node * FD + lane * 4) = o;
}

// ---------------- K5: out = Xi @ (beta*W + (1-beta)*I)^T via fp32 WMMA ----------------
// Block = 256 threads = 8 wave32; each wave owns a 16-node tile.
// W' staged to LDS in two 64-row phases, row stride 132 floats (bank-conflict-free
// b64 reads given the A/B lane layout: lanes 0-15 -> K+0/1, lanes 16-31 -> K+2/3).
#define WSTRIDE 132
__global__ __launch_bounds__(256) void gemm_out(const float* __restrict__ Xi,
                                                const float* __restrict__ W,
                                                const float* __restrict__ beta_p,
                                                float* __restrict__ out, int N_) {
    __shared__ float Wp[64 * WSTRIDE];   // 33,792 B
    const int tid = threadIdx.x;
    const float beta = *beta_p;
    const float ombeta = 1.0f - beta;
    const int lane = tid & 31;
    const int wave = tid >> 5;
    const int half = lane >> 4;      // 0: lanes 0-15, 1: lanes 16-31
    const int n16  = lane & 15;
    const int koff = half * 2;       // A/B fragment K offset per ISA 16x4 / 4x16 layout
    const int nodeBase = blockIdx.x * 128 + wave * 16;
    const bool active = (nodeBase + 16) <= N_;

    v8f acc[8];
    #pragma unroll
    for (int i = 0; i < 8; ++i) acc[i] = (v8f){};

    for (int phase = 0; phase < 2; ++phase) {
        __syncthreads();
        // stage W' = beta*W + (1-beta)*I rows [phase*64, phase*64+64)
        #pragma unroll
        for (int i = 0; i < 32; ++i) {
            int idx = i * 256 + tid;     // 8192 elements / 256 threads
            int jl  = idx >> 7;          // local row 0..63
            int k   = idx & 127;
            int j   = (phase << 6) + jl;
            float wv = beta * W[(size_t)j * FD + k];
            if (j == k) wv += ombeta;
            Wp[jl * WSTRIDE + k] = wv;
        }
        __syncthreads();
        if (active) {
            const float* arow = Xi + (size_t)(nodeBase + n16) * FD;
            for (int k = 0; k < 128; k += 4) {
                v2f a = *(const v2f*)(arow + k + koff);           // A frag: M=n16, K=k+koff..+1
                #pragma unroll
                for (int jb = 0; jb < 4; ++jb) {
                    v2f b = *(const v2f*)(&Wp[(jb * 16 + n16) * WSTRIDE + k + koff]); // B: N=n16
                    acc[phase * 4 + jb] = __builtin_amdgcn_wmma_f32_16x16x4_f32(
                        false, a, false, b, (short)0, acc[phase * 4 + jb], false, false);
                }
            }
        }
    }
    if (!active) return;
    // C/D layout: VGPR r -> row r (lanes 0-15) / row r+8 (lanes 16-31), col = n16
    #pragma unroll
    for (int r = 0; r < 8; ++r) {
        int mrow = r + half * 8;
        float* orow = out + (size_t)(nodeBase + mrow) * FD;
        #pragma unroll
        for (int jb8 = 0; jb8 < 8; ++jb8) {
            orow[jb8 * 16 + n16] = acc[jb8][r];
        }
    }
}

extern "C" void kernel_launch(void* const* d_in, const int* in_sizes, int n_in,
                              void* d_out, int out_size, void* d_ws, size_t ws_size,
                              hipStream_t stream) {
    const float* X     = (const float*)d_in[0];
    const float* X0    = (const float*)d_in[1];
    const float* degE  = (const float*)d_in[2];
    const float* degV  = (const float*)d_in[3];
    const float* W     = (const float*)d_in[4];
    const float* alpha = (const float*)d_in[5];
    const float* beta  = (const float*)d_in[6];
    const float* ln_w  = (const float*)d_in[7];
    const float* ln_b  = (const float*)d_in[8];
    const int* vertex  = (const int*)d_in[9];
    const int* edges   = (const int*)d_in[10];
    float* out = (float*)d_out;

    const int E_ = in_sizes[2];        // degE: (E,1)
    const int N_ = in_sizes[3];        // degV: (N,1)
    const int M_ = in_sizes[9];        // vertex: (M,)

    // workspace layout (floats): [esum E*D][ecnt E][Xv N*D][Xe E*D][Xi N*D]
    float* ws   = (float*)d_ws;
    float* esum = ws;
    float* ecnt = esum + (size_t)E_ * FD;
    float* Xv   = ecnt + E_;
    float* Xe   = Xv + (size_t)N_ * FD;
    float* Xi   = Xe + (size_t)E_ * FD;

    // zero the accumulated regions (esum, ecnt, Xv are contiguous from ws start)
    size_t nzero = (size_t)E_ * FD + (size_t)E_ + (size_t)N_ * FD;
    zero_kernel<<<2048, 256, 0, stream>>>(ws, nzero);

    int g1 = (M_ * 32 + 255) / 256;
    scatter_v2e<<<g1, 256, 0, stream>>>(X, vertex, edges, esum, ecnt, M_);

    int g2 = (E_ * 32 + 255) / 256;
    edge_finalize<<<g2, 256, 0, stream>>>(esum, ecnt, degE, Xe, E_);

    scatter_e2v<<<g1, 256, 0, stream>>>(Xe, vertex, edges, Xv, M_);

    int g4 = (N_ + 7) / 8;
    node_ln<<<g4, 256, 0, stream>>>(Xv, degV, X0, ln_w, ln_b, alpha, Xi, N_);

    int tiles = (N_ + 15) / 16;
    int g5 = (tiles + 7) / 8;
    gemm_out<<<g5, 256, 0, stream>>>(Xi, W, beta, out, N_);
}